// Transformer_14370960572976
// MI455X (gfx1250) — compile-verified
//
#include <hip/hip_runtime.h>
#include <math.h>

#define NBOX    1024
#define THREADS 256
#define NWAVES  (THREADS / 32)
#define NT16    (NBOX / 16)            // 64 tiles per dimension
#define SCORE_THR 0.05f
#define KEY_INF   3.0e38f

typedef __attribute__((ext_vector_type(2))) float v2f;
typedef __attribute__((ext_vector_type(8))) float v8f;

// One workgroup per (batch, class). Greedy NMS exactly matching the
// reference fori_loop semantics, with the full 1024x1024 suppression
// bit-matrix held in CDNA5's 320KB LDS and the pairwise (area_i + area_j)
// IoU denominators produced by v_wmma_f32_16x16x4_f32 tiles.
__global__ __launch_bounds__(THREADS) void nms_kernel(
    const float* __restrict__ cls,      // (B, C, A)
    const float* __restrict__ reg,      // (B, 4, A)
    const float* __restrict__ anchors,  // (A, 4)
    float* __restrict__ out,            // (B, C, A, 5)
    int B, int C, int A, float W, float H)
{
  __shared__ unsigned short mask16[NBOX][NT16];  // 128 KB suppression matrix
  __shared__ float4   bxs[NBOX];                 // 16 KB boxes (sorted order)
  __shared__ float    areas[NBOX];               // 4 KB
  __shared__ float    keys[NBOX];                // 4 KB (-score | +INF)
  __shared__ unsigned idxs[NBOX];                // 4 KB original anchor ids
  __shared__ unsigned keepw[NBOX / 32];

  const int tid  = threadIdx.x;
  const int lane = tid & 31;
  const int wid  = tid >> 5;
  const int b    = blockIdx.x / C;
  const int c    = blockIdx.x % C;
  const float* clsrow = cls + ((size_t)b * C + c) * A;

  // ---- 1) sort keys: valid -> -score (ascending = descending score), else +INF
  for (int a = tid; a < NBOX; a += THREADS) {
    float s = (a < A) ? clsrow[a] : 0.0f;
    keys[a] = (s > SCORE_THR) ? -s : KEY_INF;
    idxs[a] = (unsigned)a;
  }

  // ---- 2) bitonic sort (key, idx) ascending
  for (unsigned k = 2; k <= NBOX; k <<= 1) {
    for (unsigned j = k >> 1; j > 0; j >>= 1) {
      __syncthreads();
      for (unsigned t = tid; t < NBOX / 2; t += THREADS) {
        unsigned i = (t << 1) - (t & (j - 1));   // (i & j) == 0
        unsigned l = i + j;
        float ki = keys[i], kl = keys[l];
        bool up = ((i & k) == 0);
        if ((ki > kl) == up) {
          keys[i] = kl; keys[l] = ki;
          unsigned ti = idxs[i]; idxs[i] = idxs[l]; idxs[l] = ti;
        }
      }
    }
  }
  __syncthreads();

  // ---- 3) zero the (sub-diagonal) mask + regress/clip boxes into sorted order
  for (int t = tid; t < NBOX * NT16 / 2; t += THREADS)
    ((unsigned*)mask16)[t] = 0u;
  for (int kk = tid; kk < NBOX; kk += THREADS) {
    int a = (int)idxs[kk];
    float ax1 = anchors[a * 4 + 0], ay1 = anchors[a * 4 + 1];
    float ax2 = anchors[a * 4 + 2], ay2 = anchors[a * 4 + 3];
    float aw = ax2 - ax1, ah = ay2 - ay1;
    float cx = ax1 + 0.5f * aw, cy = ay1 + 0.5f * ah;
    float dx = reg[((size_t)b * 4 + 0) * A + a] * 0.1f;
    float dy = reg[((size_t)b * 4 + 1) * A + a] * 0.1f;
    float dw = reg[((size_t)b * 4 + 2) * A + a] * 0.2f;
    float dh = reg[((size_t)b * 4 + 3) * A + a] * 0.2f;
    float pcx = cx + dx * aw, pcy = cy + dy * ah;
    float pw = expf(dw) * aw, ph = expf(dh) * ah;
    float x1 = fmaxf(pcx - 0.5f * pw, 0.0f);
    float y1 = fmaxf(pcy - 0.5f * ph, 0.0f);
    float x2 = fminf(pcx + 0.5f * pw, W);
    float y2 = fminf(pcy + 0.5f * ph, H);
    bxs[kk]   = make_float4(x1, y1, x2, y2);
    areas[kk] = (x2 - x1) * (y2 - y1);
  }
  __syncthreads();

  // ---- 4) suppression matrix, upper-triangular 16x16 tiles only.
  // WMMA: D[i][j] = area_i*1 + 1*area_j via A=[a_i,1,0,0], B=[1,a_j,0,0]^T.
  // Tile index is wave-uniform at runtime; readfirstlane makes it scalar so
  // the jt>=it skip is an s_cbranch and WMMA always sees EXEC = all-1s.
  for (int tile = wid; tile < NT16 * NT16; tile += NWAVES) {
    int tileu = __builtin_amdgcn_readfirstlane(tile);
    int it = tileu >> 6;
    int jt = tileu & (NT16 - 1);
    if (jt >= it) {
      v2f Aop, Bop;
      if (lane < 16) {                   // rows M=0..15: K0=a_i, K1=1
        Aop.x = areas[it * 16 + lane]; Aop.y = 1.0f;
        Bop.x = 1.0f;                  Bop.y = areas[jt * 16 + lane];
      } else {                           // K2/K3 contributions must be zero
        Aop.x = 0.0f; Aop.y = 0.0f; Bop.x = 0.0f; Bop.y = 0.0f;
      }
      v8f Cz = {};
      v8f S = __builtin_amdgcn_wmma_f32_16x16x4_f32(
          false, Aop, false, Bop, (short)0, Cz, false, false);

      int jj = jt * 16 + (lane & 15);
      float4 bj = bxs[jj];
      int ibase = it * 16 + ((lane >> 4) << 3);  // lanes 16-31 do rows +8
#pragma unroll
      for (int v = 0; v < 8; ++v) {
        int ii = ibase + v;
        float4 bi = bxs[ii];
        float ix1 = fmaxf(bi.x, bj.x), iy1 = fmaxf(bi.y, bj.y);
        float ix2 = fminf(bi.z, bj.z), iy2 = fminf(bi.w, bj.w);
        float inter = fmaxf(ix2 - ix1, 0.0f) * fmaxf(iy2 - iy1, 0.0f);
        // iou > 0.5  <=>  3*inter > area_i + area_j   (S[v] from WMMA)
        bool pred = (3.0f * inter > S[v]) && (jj > ii);
        unsigned bal = __builtin_amdgcn_ballot_w32(pred);
        if (lane == 0)  mask16[it * 16 + v    ][jt] = (unsigned short)(bal & 0xFFFFu);
        if (lane == 16) mask16[it * 16 + v + 8][jt] = (unsigned short)(bal >> 16);
      }
    }
  }
  __syncthreads();

  // ---- 5) sequential greedy scan, single wave, lane w owns keep word w
  if (tid < 32) {
    unsigned kw = 0;
    for (int t = 0; t < 32; ++t)
      if (keys[tid * 32 + t] < KEY_INF) kw |= (1u << t);   // keep0 = valid
    for (int i = 0; i < NBOX; ++i) {
      unsigned cur = __shfl(kw, i >> 5, 32);
      if ((cur >> (i & 31)) & 1u) {                        // keep[i] still set
        const unsigned* mrow = (const unsigned*)&mask16[i][0];
        kw &= ~mrow[tid];                                  // mask has only j>i bits
      }
    }
    keepw[tid] = kw;
  }
  __syncthreads();

  // ---- 6) stage output in LDS (reusing the dead mask buffer), then stream
  // the (A,5) row to HBM with fully coalesced contiguous stores.
  float* outl = (float*)mask16;   // 20 KB needed, 128 KB available
  for (int kk = tid; kk < NBOX; kk += THREADS) {
    int a = (int)idxs[kk];
    if (a >= A) continue;
    bool kp = (keepw[kk >> 5] >> (kk & 31)) & 1u;
    float4 bb = bxs[kk];
    float* p = outl + (size_t)a * 5;
    p[0] = kp ? -keys[kk] : 0.0f;
    p[1] = kp ? bb.x : 0.0f;
    p[2] = kp ? bb.y : 0.0f;
    p[3] = kp ? bb.z : 0.0f;
    p[4] = kp ? bb.w : 0.0f;
  }
  __syncthreads();
  float* orow = out + ((size_t)b * C + c) * (size_t)A * 5;
  for (int t = tid; t < A * 5; t += THREADS)
    orow[t] = outl[t];
}

extern "C" void kernel_launch(void* const* d_in, const int* in_sizes, int n_in,
                              void* d_out, int out_size, void* d_ws, size_t ws_size,
                              hipStream_t stream) {
  // inputs: image (B,3,H,W) [shape only], cls_pred (B,C,A), reg_pred (B,4,A), anchors (A,4)
  const float* cls = (const float*)d_in[1];
  const float* reg = (const float*)d_in[2];
  const float* anc = (const float*)d_in[3];
  float* out = (float*)d_out;

  int A = in_sizes[3] / 4;
  int B = in_sizes[2] / (4 * A);
  int C = in_sizes[1] / (B * A);
  long hw = (long)in_sizes[0] / (3L * B);
  int H = (int)(sqrt((double)hw) + 0.5);
  int W = H;

  dim3 grid(B * C), block(THREADS);
  hipLaunchKernelGGL(nms_kernel, grid, block, 0, stream,
                     cls, reg, anc, out, B, C, A, (float)W, (float)H);
}